// GraphNet_7876970020890
// MI455X (gfx1250) — compile-verified
//
#include <hip/hip_runtime.h>
#include <math.h>

#define F_IN 512
#define HID  16
#define NC   40
#define XS_STRIDE 68   // 64 K-chunk + 4 dword pad -> bank = (4m+k)%64, conflict-free

typedef float v2f __attribute__((ext_vector_type(2)));
typedef float v8f __attribute__((ext_vector_type(8)));
typedef unsigned int u32x4 __attribute__((ext_vector_type(4)));
typedef int i32x4 __attribute__((ext_vector_type(4)));
typedef int i32x8 __attribute__((ext_vector_type(8)));

static __device__ __forceinline__ void atomicAddF(float* p, float v) {
    __hip_atomic_fetch_add(p, v, __ATOMIC_RELAXED, __HIP_MEMORY_SCOPE_AGENT);
}

// ---------------- init kernels ----------------
__global__ void k_set1(float* p, int n) {
    int i = blockIdx.x * blockDim.x + threadIdx.x;
    if (i < n) p[i] = 1.0f;   // self-loop weight pre-seeded into degree
}
__global__ void k_zero(float* p, int n) {
    int i = blockIdx.x * blockDim.x + threadIdx.x;
    if (i < n) p[i] = 0.0f;
}

// ---------------- degree accumulate ----------------
__global__ void k_degacc(float* deg, const int* __restrict__ ei,
                         const float* __restrict__ w, int E) {
    int e = blockIdx.x * blockDim.x + threadIdx.x;
    if (e < E) atomicAddF(&deg[ei[e]], w[e]);
}

// deg -> d^{-1/2} in place
__global__ void k_dinv(float* deg, int n) {
    int i = blockIdx.x * blockDim.x + threadIdx.x;
    if (i < n) {
        float d = deg[i];
        deg[i] = (d > 0.0f) ? rsqrtf(d) : 0.0f;
    }
}

// norm[e] = dinv[row]*w*dinv[col]  (self-loops appended at e in [E, E+N))
__global__ void k_norm(float* __restrict__ nrm, const int* __restrict__ ei,
                       const float* __restrict__ w, const float* __restrict__ dinv,
                       int E, int N) {
    int e = blockIdx.x * blockDim.x + threadIdx.x;
    if (e >= E + N) return;
    if (e < E) {
        int r = ei[e], c = ei[E + e];
        nrm[e] = dinv[r] * w[e] * dinv[c];
    } else {
        float d = dinv[e - E];
        nrm[e] = d * d;     // self-loop, weight 1
    }
}

// ---------------- GEMM1: h1 = x @ W1  (N x 512 @ 512 x 16), fp32 WMMA ----------------
// x tile staged HBM -> LDS by the Tensor Data Mover (128 rows x 64 cols per chunk,
// LDS padding 4 dwords per 64 -> row stride 68), W fragments from LDS, fp32 WMMA.
__global__ __launch_bounds__(256) void k_gemm1(const float* __restrict__ x,
                                               const float* __restrict__ W1,
                                               float* __restrict__ h1, int nrows) {
    __shared__ float xs[128 * XS_STRIDE];
    __shared__ float wl[64 * 16];    // W1 chunk, k-major
    const int t    = threadIdx.x;
    const int wv   = t >> 5;         // wave id 0..7, wave32
    const int lane = t & 31;
    const int m    = lane & 15;
    const int hi   = lane >> 4;      // lane half selects K pair {0,1} vs {2,3}
    const int rowBase = blockIdx.x * 128;
#if __has_builtin(__builtin_amdgcn_tensor_load_to_lds)
    const unsigned lds_xs = (unsigned)(size_t)(void*)xs;
    const int wuni = __builtin_amdgcn_readfirstlane(wv);   // scalar branch gate
#endif

    v8f c = {};
    for (int kc = 0; kc < F_IN; kc += 64) {
        __syncthreads();   // previous chunk fully consumed
        // W1 chunk (tiny, regular path)
        for (int j = t; j < 64 * 16; j += 256)
            wl[j] = W1[(size_t)(kc + (j >> 4)) * HID + (j & 15)];
#if __has_builtin(__builtin_amdgcn_tensor_load_to_lds)
        if (wuni == 0) {
            // --- Tensor DMA descriptor (D#) ---
            unsigned long long ga =
                (unsigned long long)(size_t)(x + (size_t)rowBase * F_IN + kc);
            u32x4 g0;
            g0.x = 1u;                                   // count=1, user mode
            g0.y = lds_xs;                               // lds_addr (bytes)
            g0.z = (unsigned)(ga & 0xffffffffu);         // global_addr[31:0]
            g0.w = (unsigned)((ga >> 32) & 0x01ffffffu)  // global_addr[56:32]
                 | (2u << 30);                           // type = 2 ("image")
            unsigned td0 = 64u;                          // tensor_dim0 (cols this chunk)
            unsigned td1 = (unsigned)(nrows - rowBase);  // valid rows; OOB rows -> 0
            i32x8 g1;
            g1[0] = (2 << 16)      // data_size = 4B
                  | (1 << 20)      // pad_enable
                  | (5 << 22)      // pad_interval code 5 -> every 64 dwords
                  | (3 << 25);     // pad_amount   code 3 -> 4 dwords
            g1[1] = (int)((td0 & 0xffffu) << 16);                       // td0[15:0]
            g1[2] = (int)((td0 >> 16) | ((td1 & 0xffffu) << 16));       // td0[31:16], td1[15:0]
            g1[3] = (int)((td1 >> 16) | (64u << 16));                   // td1[31:16], tile_dim0=64
            g1[4] = (int)128;                                           // tile_dim1=128, tile_dim2=0
            g1[5] = (int)F_IN;                                          // tensor_dim0_stride = 512
            g1[6] = 0;
            g1[7] = 0;
            i32x4 z4 = {0, 0, 0, 0};
            i32x8 z8 = {0, 0, 0, 0, 0, 0, 0, 0};
            __builtin_amdgcn_tensor_load_to_lds(g0, g1, z4, z4, z8, 0);
            __builtin_amdgcn_s_wait_tensorcnt(0);
        }
#else
        for (int j = t; j < 128 * 64; j += 256) {
            int r = j >> 6, cc = j & 63;
            int gr = rowBase + r;
            xs[r * XS_STRIDE + cc] = (gr < nrows) ? x[(size_t)gr * F_IN + kc + cc] : 0.0f;
        }
#endif
        __syncthreads();   // tile visible to all waves

        const float* xrow = &xs[(wv * 16 + m) * XS_STRIDE];
        #pragma unroll
        for (int kk = 0; kk < 64; kk += 4) {
            const int ka = kk + 2 * hi;
            v2f a, b;
            a.x = xrow[ka];
            a.y = xrow[ka + 1];
            b.x = wl[ka * 16 + m];          // B: row K=ka, col N=m
            b.y = wl[(ka + 1) * 16 + m];
            c = __builtin_amdgcn_wmma_f32_16x16x4_f32(
                    false, a, false, b, (short)0, c, false, false);
        }
    }
    // D layout: reg r, lane -> (M = r + 8*hi, N = m)
    #pragma unroll
    for (int r = 0; r < 8; ++r) {
        int gr = rowBase + wv * 16 + r + 8 * hi;
        if (gr < nrows) h1[(size_t)gr * HID + m] = c[r];
    }
}

// ---------------- edge pass 1: agg1 += norm * h1[col], 16 feat ----------------
__global__ void k_mp1(float* agg, const float* __restrict__ h,
                      const int* __restrict__ ei, const float* __restrict__ nrm,
                      int E, int N) {
    unsigned t = blockIdx.x * blockDim.x + threadIdx.x;
    unsigned total = (unsigned)(E + N) * 16u;
    if (t >= total) return;
    int e = (int)(t >> 4), f = (int)(t & 15u);
    int r, cl;
    if (e < E) { r = ei[e]; cl = ei[E + e]; }
    else       { r = cl = e - E; }
    float v = nrm[e] * h[(size_t)cl * HID + f];
    atomicAddF(&agg[(size_t)r * HID + f], v);
}

// ---------------- GEMM2: outl = relu(agg1+b1) @ W2  (N x 16 @ 16 x 40) ----------------
__global__ __launch_bounds__(256) void k_gemm2(const float* __restrict__ agg1,
                                               const float* __restrict__ b1,
                                               const float* __restrict__ W2,
                                               float* __restrict__ outl, int nrows) {
    const int t    = threadIdx.x;
    const int wv   = t >> 5;
    const int lane = t & 31;
    const int m    = lane & 15;
    const int hi   = lane >> 4;
    const int rowBase = blockIdx.x * 128 + wv * 16;

    // A fragments for 4 K-steps, with fused bias + ReLU
    v2f a[4];
    #pragma unroll
    for (int ks = 0; ks < 4; ++ks) {
        const int kb = ks * 4 + 2 * hi;
        const int gr = rowBase + m;
        float x0 = 0.0f, x1 = 0.0f;
        if (gr < nrows) {
            x0 = fmaxf(agg1[(size_t)gr * HID + kb]     + b1[kb],     0.0f);
            x1 = fmaxf(agg1[(size_t)gr * HID + kb + 1] + b1[kb + 1], 0.0f);
        }
        a[ks].x = x0; a[ks].y = x1;
    }

    #pragma unroll
    for (int nt = 0; nt < 3; ++nt) {          // N = 40 -> 3 tiles of 16 (padded)
        const int col = nt * 16 + m;
        v8f c = {};
        #pragma unroll
        for (int ks = 0; ks < 4; ++ks) {
            const int kb = ks * 4 + 2 * hi;
            v2f b;
            b.x = (col < NC) ? W2[kb * NC + col]       : 0.0f;
            b.y = (col < NC) ? W2[(kb + 1) * NC + col] : 0.0f;
            c = __builtin_amdgcn_wmma_f32_16x16x4_f32(
                    false, a[ks], false, b, (short)0, c, false, false);
        }
        #pragma unroll
        for (int r = 0; r < 8; ++r) {
            int gr = rowBase + r + 8 * hi;
            if (gr < nrows && col < NC) outl[(size_t)gr * NC + col] = c[r];
        }
    }
}

// ---------------- edge pass 2: agg2 += norm * outl[col], 40 feat ----------------
__global__ void k_mp2(float* agg, const float* __restrict__ h,
                      const int* __restrict__ ei, const float* __restrict__ nrm,
                      int E, int N) {
    unsigned t = blockIdx.x * blockDim.x + threadIdx.x;
    unsigned total = (unsigned)(E + N) * (unsigned)NC;
    if (t >= total) return;
    int e = (int)(t / (unsigned)NC), f = (int)(t % (unsigned)NC);
    int r, cl;
    if (e < E) { r = ei[e]; cl = ei[E + e]; }
    else       { r = cl = e - E; }
    float v = nrm[e] * h[(size_t)cl * NC + f];
    atomicAddF(&agg[(size_t)r * NC + f], v);
}

// ---------------- log-softmax(agg2 + b2) ----------------
__global__ void k_lsm(float* __restrict__ out, const float* __restrict__ agg2,
                      const float* __restrict__ b2, int N) {
    int i = blockIdx.x * blockDim.x + threadIdx.x;
    if (i >= N) return;
    float z[NC];
    float mx = -INFINITY;
    #pragma unroll
    for (int c = 0; c < NC; ++c) {
        z[c] = agg2[(size_t)i * NC + c] + b2[c];
        mx = fmaxf(mx, z[c]);
    }
    float s = 0.0f;
    #pragma unroll
    for (int c = 0; c < NC; ++c) s += __expf(z[c] - mx);
    float lse = mx + __logf(s);
    #pragma unroll
    for (int c = 0; c < NC; ++c) out[(size_t)i * NC + c] = z[c] - lse;
}

extern "C" void kernel_launch(void* const* d_in, const int* in_sizes, int n_in,
                              void* d_out, int out_size, void* d_ws, size_t ws_size,
                              hipStream_t stream) {
    const float* x  = (const float*)d_in[0];
    const int*   ei = (const int*)  d_in[1];
    const float* ew = (const float*)d_in[2];
    const float* W1 = (const float*)d_in[3];
    const float* b1 = (const float*)d_in[4];
    const float* W2 = (const float*)d_in[5];
    const float* b2 = (const float*)d_in[6];

    const int N = in_sizes[0] / F_IN;
    const int E = in_sizes[2];

    // workspace layout (floats)
    float* ws   = (float*)d_ws;
    float* deg  = ws;                              // N      (becomes dinv)
    float* nrm  = deg  + N;                        // E + N
    float* h1   = nrm  + (size_t)(E + N);          // N * 16
    float* agg1 = h1   + (size_t)N * HID;          // N * 16
    float* outl = agg1 + (size_t)N * HID;          // N * 40
    float* agg2 = outl + (size_t)N * NC;           // N * 40

    const int B = 256;
    k_set1<<<(N + B - 1) / B, B, 0, stream>>>(deg, N);
    k_zero<<<(N * HID + B - 1) / B, B, 0, stream>>>(agg1, N * HID);
    k_zero<<<(N * NC + B - 1) / B, B, 0, stream>>>(agg2, N * NC);

    k_degacc<<<(E + B - 1) / B, B, 0, stream>>>(deg, ei, ew, E);
    k_dinv<<<(N + B - 1) / B, B, 0, stream>>>(deg, N);
    k_norm<<<(E + N + B - 1) / B, B, 0, stream>>>(nrm, ei, ew, deg, E, N);

    k_gemm1<<<(N + 127) / 128, 256, 0, stream>>>(x, W1, h1, N);

    {
        unsigned total = (unsigned)(E + N) * 16u;
        k_mp1<<<(total + B - 1) / B, B, 0, stream>>>(agg1, h1, ei, nrm, E, N);
    }

    k_gemm2<<<(N + 127) / 128, 256, 0, stream>>>(agg1, b1, W2, outl, N);

    {
        unsigned total = (unsigned)(E + N) * (unsigned)NC;
        k_mp2<<<(total + B - 1) / B, B, 0, stream>>>(agg2, outl, ei, nrm, E, N);
    }

    k_lsm<<<(N + B - 1) / B, B, 0, stream>>>((float*)d_out, agg2, b2, N);

    (void)n_in; (void)out_size; (void)ws_size;
}